// GNNModel_33088428048433
// MI455X (gfx1250) — compile-verified
//
#include <hip/hip_runtime.h>
#include <math.h>

#define HEADS 4
#define N_ATOMS 64
#define NEG_SLOPE 0.2f
#define JK_DIM 1248

#define KC 64          // K chunk staged in LDS
#define BSTR 66        // padded LDS column stride (floats): conflict-free b64 reads

typedef float v2f __attribute__((ext_vector_type(2)));
typedef float v8f __attribute__((ext_vector_type(8)));

// ---------------------------------------------------------------------------
// WMMA GEMM: C[M,Nc] = A[M,K] @ B[K,Nc] (+ bias[Nc] if non-null)
// Block = 256 threads = 8 waves; block computes a 128x32 output region.
//  - B panel (KCx32) staged in LDS col-major (stride 66), double-buffered,
//    one barrier per chunk; each wave reads its fragments as ds_load_b64.
//  - Each wave holds two 16x16 accumulators (16x32 strip) and reuses the A
//    fragment across two consecutive WMMAs.
//  - Inner K-loop unrolled 4x (16 K / iter) so load clauses run ahead of the
//    matrix pipe.
// Requires M % 128 == 0, Nc % 32 == 0, K % 16 == 0 (all true here).
// ---------------------------------------------------------------------------
__global__ void __launch_bounds__(256)
wmma_gemm_f32(const float* __restrict__ A, const float* __restrict__ B,
              const float* __restrict__ bias, float* __restrict__ C,
              int M, int K, int Nc) {
  __shared__ float Bs[2][32 * BSTR];

  const int tid  = (int)threadIdx.x;
  const int wave = tid >> 5;
  const int lane = tid & 31;
  const int colBlocks = Nc >> 5;
  const int bm = (int)(blockIdx.x / colBlocks) << 7;   // 128-row block base
  const int bn = (int)(blockIdx.x % colBlocks) << 5;   // 32-col block base
  const int tm = bm + (wave << 4);

  const int rowA = tm + (lane & 15);
  const int kh   = (lane >> 4) << 1;                   // 0 or 2 (upper K half)
  const int cc   = lane & 15;

  v8f acc0 = {}, acc1 = {};
  const float* ap = A + (size_t)rowA * K + kh;

  int buf = 0;
  for (int k0 = 0; k0 < K; k0 += KC) {
    const int kc = (K - k0 < KC) ? (K - k0) : KC;
    // cooperative load: B[k0..k0+kc) x [bn..bn+32) -> Bs[buf], col-major
    for (int i = tid; i < (kc << 5); i += 256) {
      const int c = i & 31, r = i >> 5;
      Bs[buf][c * BSTR + r] = B[(size_t)(k0 + r) * Nc + bn + c];
    }
    __syncthreads();
    const float* bs0 = &Bs[buf][cc * BSTR + kh];
    const float* bs1 = &Bs[buf][(16 + cc) * BSTR + kh];
    for (int k = 0; k < kc; k += 16) {
      v2f a0 = *(const v2f*)(ap);
      v2f a1 = *(const v2f*)(ap + 4);
      v2f a2 = *(const v2f*)(ap + 8);
      v2f a3 = *(const v2f*)(ap + 12);
      v2f p0 = *(const v2f*)(bs0 + k);
      v2f q0 = *(const v2f*)(bs1 + k);
      v2f p1 = *(const v2f*)(bs0 + k + 4);
      v2f q1 = *(const v2f*)(bs1 + k + 4);
      v2f p2 = *(const v2f*)(bs0 + k + 8);
      v2f q2 = *(const v2f*)(bs1 + k + 8);
      v2f p3 = *(const v2f*)(bs0 + k + 12);
      v2f q3 = *(const v2f*)(bs1 + k + 12);
      ap += 16;
      acc0 = __builtin_amdgcn_wmma_f32_16x16x4_f32(
          false, a0, false, p0, (short)0, acc0, false, false);
      acc1 = __builtin_amdgcn_wmma_f32_16x16x4_f32(
          false, a0, false, q0, (short)0, acc1, false, false);
      acc0 = __builtin_amdgcn_wmma_f32_16x16x4_f32(
          false, a1, false, p1, (short)0, acc0, false, false);
      acc1 = __builtin_amdgcn_wmma_f32_16x16x4_f32(
          false, a1, false, q1, (short)0, acc1, false, false);
      acc0 = __builtin_amdgcn_wmma_f32_16x16x4_f32(
          false, a2, false, p2, (short)0, acc0, false, false);
      acc1 = __builtin_amdgcn_wmma_f32_16x16x4_f32(
          false, a2, false, q2, (short)0, acc1, false, false);
      acc0 = __builtin_amdgcn_wmma_f32_16x16x4_f32(
          false, a3, false, p3, (short)0, acc0, false, false);
      acc1 = __builtin_amdgcn_wmma_f32_16x16x4_f32(
          false, a3, false, q3, (short)0, acc1, false, false);
    }
    buf ^= 1;   // double buffer: next load targets the other panel
  }

  // C/D layout: VGPR i -> row tm+i (lanes 0-15) / tm+8+i (lanes 16-31)
  const int r0   = tm + ((lane >> 4) << 3);
  const int col0 = bn + cc;
  const float bv0 = bias ? bias[col0] : 0.f;
  const float bv1 = bias ? bias[col0 + 16] : 0.f;
#pragma unroll
  for (int i = 0; i < 8; ++i) {
    C[(size_t)(r0 + i) * Nc + col0]      = acc0[i] + bv0;
    C[(size_t)(r0 + i) * Nc + col0 + 16] = acc1[i] + bv1;
  }
}

// ---------------------------------------------------------------------------
// Attention scores per (node, head) + init of segment max/denominator.
// ---------------------------------------------------------------------------
__global__ void attn_scores(const float* __restrict__ h,
                            const float* __restrict__ att_src,
                            const float* __restrict__ att_dst,
                            float* __restrict__ a_s, float* __restrict__ a_d,
                            float* __restrict__ m,  float* __restrict__ z,
                            int n, int C) {
  int t = blockIdx.x * blockDim.x + threadIdx.x;
  if (t >= n * HEADS) return;
  int node = t / HEADS, head = t % HEADS;
  const float* hp = h + (size_t)node * HEADS * C + (size_t)head * C;
  const float* as = att_src + head * C;
  const float* ad = att_dst + head * C;
  float s = 0.f, d = 0.f;
  for (int c = 0; c < C; ++c) { float v = hp[c]; s += v * as[c]; d += v * ad[c]; }
  a_s[t] = s;
  a_d[t] = d;
  m[t]   = __int_as_float(0xFF800000);   // -inf
  z[t]   = 0.f;
}

// Sign-aware float atomic max (works with -inf init).
__device__ __forceinline__ void atomicMaxFloat(float* addr, float val) {
  if (val >= 0.f)
    atomicMax((int*)addr, __float_as_int(val));
  else
    atomicMin((unsigned int*)addr, __float_as_uint(val));
}

// pass 1: e = leaky_relu(a_s[src]+a_d[dst]); segment max over dst
__global__ void edge_pass1(const int* __restrict__ src, const int* __restrict__ dst,
                           const float* __restrict__ a_s, const float* __restrict__ a_d,
                           float* __restrict__ e, float* __restrict__ m,
                           int E, int n) {
  int t = blockIdx.x * blockDim.x + threadIdx.x;
  int total = (E + n) * HEADS;
  if (t >= total) return;
  int eid = t / HEADS, head = t % HEADS;
  int s, d;
  if (eid < E) { s = src[eid]; d = dst[eid]; } else { s = d = eid - E; }
  float v = a_s[s * HEADS + head] + a_d[d * HEADS + head];
  v = (v > 0.f) ? v : NEG_SLOPE * v;
  e[t] = v;
  atomicMaxFloat(&m[d * HEADS + head], v);
}

// pass 2: w = exp(e - m[dst]) (stored in place of e); z[dst] += w
__global__ void edge_pass2(const int* __restrict__ src, const int* __restrict__ dst,
                           float* __restrict__ e, const float* __restrict__ m,
                           float* __restrict__ z, int E, int n) {
  int t = blockIdx.x * blockDim.x + threadIdx.x;
  int total = (E + n) * HEADS;
  if (t >= total) return;
  int eid = t / HEADS, head = t % HEADS;
  int d = (eid < E) ? dst[eid] : (eid - E);
  float w = __expf(e[t] - m[d * HEADS + head]);
  e[t] = w;
  atomicAdd(&z[d * HEADS + head], w);
}

__global__ void zero_f32(float* __restrict__ p, int n) {
  int t = blockIdx.x * blockDim.x + threadIdx.x;
  if (t < n) p[t] = 0.f;
}

// pass 3: out[dst] += h[src] * alpha  (thread per edge x 4-feature group)
__global__ void edge_pass3(const int* __restrict__ src, const int* __restrict__ dst,
                           const float* __restrict__ w, const float* __restrict__ z,
                           const float* __restrict__ h, float* __restrict__ out,
                           int E, int n, int outc) {
  int fg = outc >> 2;
  int t = blockIdx.x * blockDim.x + threadIdx.x;
  int total = (E + n) * fg;
  if (t >= total) return;
  int eid = t / fg, g = t % fg;
  int s, d;
  if (eid < E) { s = src[eid]; d = dst[eid]; } else { s = d = eid - E; }
  int C = outc / HEADS;
  int head = (g << 2) / C;                       // C % 4 == 0 -> no head split
  float alpha = w[eid * HEADS + head] /
                (z[d * HEADS + head] + 1e-16f);
  const float4 hv = *(const float4*)(h + (size_t)s * outc + (g << 2));
  float* op = out + (size_t)d * outc + (g << 2);
  atomicAdd(op + 0, hv.x * alpha);
  atomicAdd(op + 1, hv.y * alpha);
  atomicAdd(op + 2, hv.z * alpha);
  atomicAdd(op + 3, hv.w * alpha);
}

// epilogue: x_next = relu(out + bias) + x_res
__global__ void layer_finish(const float* __restrict__ out,
                             const float* __restrict__ bias,
                             const float* __restrict__ xres,
                             float* __restrict__ xnext, int n, int outc) {
  int t = blockIdx.x * blockDim.x + threadIdx.x;
  if (t >= n * outc) return;
  float v = out[t] + bias[t % outc];
  v = (v > 0.f) ? v : 0.f;
  xnext[t] = v + xres[t];
}

// per-graph mean pool of this layer's output into the JK buffer slice
__global__ void pool_accum(const float* __restrict__ x, float* __restrict__ jk,
                           int outc, int jkoff) {
  int g = blockIdx.x;
  const float* xp = x + (size_t)g * N_ATOMS * outc;
  for (int f = threadIdx.x; f < outc; f += blockDim.x) {
    float s = 0.f;
    for (int a = 0; a < N_ATOMS; ++a) s += xp[(size_t)a * outc + f];
    jk[(size_t)g * JK_DIM + jkoff + f] = s * (1.f / (float)N_ATOMS);
  }
}

// ---------------------------------------------------------------------------
extern "C" void kernel_launch(void* const* d_in, const int* in_sizes, int n_in,
                              void* d_out, int out_size, void* d_ws, size_t ws_size,
                              hipStream_t stream) {
  (void)n_in; (void)out_size; (void)ws_size;

  const float* x0 = (const float*)d_in[0];
  const int*   ei = (const int*)d_in[1];
  const int E = in_sizes[1] / 2;
  const int n = in_sizes[0] / 128;
  const int G = n / N_ATOMS;
  const int ET = E + n;
  const int* src = ei;
  const int* dst = ei + E;

  static const int IN_C[8]   = {128, 32, 64, 128, 128, 128, 256, 256};
  static const int OUT_C[8]  = { 32, 64, 128, 128, 128, 256, 256, 256};
  static const int HASRES[8] = {  1,  1,   1,   0,   0,   1,   0,   0};

  // Leaf order: x, edge_index, batch, then per layer {W, att_src, att_dst,
  // bias, [res_W, res_b]}, then fc_w, fc_b  (insertion-order flattening,
  // None skipped).
  int pW[8], pAS[8], pAD[8], pB[8], pRW[8], pRB[8];
  int idx = 3;
  for (int l = 0; l < 8; ++l) {
    pW[l] = idx++; pAS[l] = idx++; pAD[l] = idx++; pB[l] = idx++;
    if (HASRES[l]) { pRW[l] = idx++; pRB[l] = idx++; }
    else           { pRW[l] = -1;   pRB[l] = -1;    }
  }
  const int pFCW = idx++, pFCB = idx++;

  // workspace partition
  char* wsp = (char*)d_ws;
  auto take = [&](size_t bytes) {
    char* p = wsp;
    wsp += (bytes + 255) & ~(size_t)255;
    return p;
  };
  float* bufA = (float*)take((size_t)n * 256 * 4);
  float* bufB = (float*)take((size_t)n * 256 * 4);
  float* hbuf = (float*)take((size_t)n * 256 * 4);
  float* xres = (float*)take((size_t)n * 256 * 4);
  float* oacc = (float*)take((size_t)n * 256 * 4);
  float* a_s  = (float*)take((size_t)n * HEADS * 4);
  float* a_d  = (float*)take((size_t)n * HEADS * 4);
  float* mbuf = (float*)take((size_t)n * HEADS * 4);
  float* zbuf = (float*)take((size_t)n * HEADS * 4);
  float* ew   = (float*)take((size_t)ET * HEADS * 4);
  float* jk   = (float*)take((size_t)G * JK_DIM * 4);

  const float* xcur = x0;
  float* xnext = bufA;
  int jkoff = 0;

  for (int l = 0; l < 8; ++l) {
    const int inc = IN_C[l], outc = OUT_C[l], C = outc / HEADS;
    const float* W   = (const float*)d_in[pW[l]];
    const float* aS  = (const float*)d_in[pAS[l]];
    const float* aD  = (const float*)d_in[pAD[l]];
    const float* bia = (const float*)d_in[pB[l]];

    {   // h = xcur @ W
      int blocks = (n / 128) * (outc / 32);
      wmma_gemm_f32<<<blocks, 256, 0, stream>>>(xcur, W, nullptr, hbuf, n, inc, outc);
    }

    const float* xr;
    if (HASRES[l]) {   // x_res = xcur @ res_W + res_b
      int blocks = (n / 128) * (outc / 32);
      wmma_gemm_f32<<<blocks, 256, 0, stream>>>(
          xcur, (const float*)d_in[pRW[l]], (const float*)d_in[pRB[l]],
          xres, n, inc, outc);
      xr = xres;
    } else {
      xr = xcur;
    }

    attn_scores<<<(n * HEADS + 255) / 256, 256, 0, stream>>>(
        hbuf, aS, aD, a_s, a_d, mbuf, zbuf, n, C);

    edge_pass1<<<(ET * HEADS + 255) / 256, 256, 0, stream>>>(
        src, dst, a_s, a_d, ew, mbuf, E, n);
    edge_pass2<<<(ET * HEADS + 255) / 256, 256, 0, stream>>>(
        src, dst, ew, mbuf, zbuf, E, n);

    zero_f32<<<(n * outc + 255) / 256, 256, 0, stream>>>(oacc, n * outc);

    {
      int fg = outc >> 2;
      int total = ET * fg;
      edge_pass3<<<(total + 255) / 256, 256, 0, stream>>>(
          src, dst, ew, zbuf, hbuf, oacc, E, n, outc);
    }

    layer_finish<<<(n * outc + 255) / 256, 256, 0, stream>>>(
        oacc, bia, xr, xnext, n, outc);

    pool_accum<<<G, 256, 0, stream>>>(xnext, jk, outc, jkoff);
    jkoff += outc;

    xcur  = xnext;
    xnext = (xnext == bufA) ? bufB : bufA;
  }

  {   // final FC: d_out[G,192] = jk @ fc_w + fc_b   (jk already mean-pooled)
    int blocks = (G / 128) * (192 / 32);
    wmma_gemm_f32<<<blocks, 256, 0, stream>>>(
        jk, (const float*)d_in[pFCW], (const float*)d_in[pFCB],
        (float*)d_out, G, JK_DIM, 192);
  }
}